// ParticleExplorer_64561948393590
// MI455X (gfx1250) — compile-verified
//
#include <hip/hip_runtime.h>
#include <cstdint>
#include <math.h>

// ---------------------------------------------------------------------------
// MI455X / gfx1250. Recurrence re-associated into quaternion prefix-product +
// float3 prefix-sum so the kernel is memory-bound (~433MB -> ~19us @23.3TB/s)
// instead of latency-bound. Async global->LDS (CDNA5 ASYNCcnt path) used for
// input staging, nontemporal b128 stores for the streamed X output.
// ---------------------------------------------------------------------------

#define AS1 __attribute__((address_space(1)))
#define AS3 __attribute__((address_space(3)))

#if defined(__has_builtin)
#  if __has_builtin(__builtin_amdgcn_global_load_async_to_lds_b128)
#    define HAVE_ASYNC 1
#  endif
#endif
#ifndef HAVE_ASYNC
#  define HAVE_ASYNC 0
#endif

#if HAVE_ASYNC
#  if __has_builtin(__builtin_amdgcn_s_wait_asynccnt)
#    define WAIT_ASYNC0() __builtin_amdgcn_s_wait_asynccnt(0)
#  else
#    define WAIT_ASYNC0() asm volatile("s_wait_asynccnt 0" ::: "memory")
#  endif
#else
#  define WAIT_ASYNC0() do {} while (0)
#endif

typedef float v4f __attribute__((ext_vector_type(4)));
typedef int   v4i __attribute__((ext_vector_type(4)));

struct Quat { float w, x, y, z; };
struct F3   { float x, y, z; };

__device__ __forceinline__ Quat qmul(const Quat& a, const Quat& b) {
  Quat r;
  r.w = a.w*b.w - a.x*b.x - a.y*b.y - a.z*b.z;
  r.x = a.w*b.x + a.x*b.w + a.y*b.z - a.z*b.y;
  r.y = a.w*b.y - a.x*b.z + a.y*b.w + a.z*b.x;
  r.z = a.w*b.z + a.x*b.y - a.y*b.x + a.z*b.w;
  return r;
}

__device__ __forceinline__ Quat qshfl_up(const Quat& q, int d) {
  Quat r;
  r.w = __shfl_up(q.w, d, 32);
  r.x = __shfl_up(q.x, d, 32);
  r.y = __shfl_up(q.y, d, 32);
  r.z = __shfl_up(q.z, d, 32);
  return r;
}

__device__ __forceinline__ F3 fshfl_up(const F3& v, int d) {
  F3 r;
  r.x = __shfl_up(v.x, d, 32);
  r.y = __shfl_up(v.y, d, 32);
  r.z = __shfl_up(v.z, d, 32);
  return r;
}

// ---------------------------------------------------------------------------
// ts + zero fill for the "states" output.
// ---------------------------------------------------------------------------
__global__ void ParticleExplorer_64561948393590_misc(float* __restrict__ ts,
                                                     float* __restrict__ states,
                                                     int nsteps, long long total)
{
  long long i0     = (long long)blockIdx.x * blockDim.x + threadIdx.x;
  long long stride = (long long)gridDim.x * blockDim.x;
  for (long long i = i0; i < total; i += stride) states[i] = 0.0f;
  for (long long i = i0; i < nsteps; i += stride) ts[i] = (float)i;  // dt == 1
}

// ---------------------------------------------------------------------------
// One block per particle. 256 threads x 4 steps = N = 1024 steps.
// F_t = F_0 * prod_{k<=t} Rz(pa_k)*Ry(na_k)   (body-axis rotations compose
// on the right; per-step normalise/orthogonalise are exact-math identities).
// x_t = x_0 + sum_{k<=t} sp_k * e0_k.
// ---------------------------------------------------------------------------
__global__ __launch_bounds__(256)
void ParticleExplorer_64561948393590_scan(const float* __restrict__ x0,
                                          const float* __restrict__ e0i,
                                          const float* __restrict__ e1i,
                                          const float* __restrict__ e2i,
                                          const float* __restrict__ sp,
                                          const float* __restrict__ pa,
                                          const float* __restrict__ na,
                                          float* __restrict__ X, int N)
{
  const int p    = blockIdx.x;
  const int t    = threadIdx.x;
  const int lane = t & 31;
  const int wid  = t >> 5;

  __shared__ float s_sp[1024], s_pa[1024], s_na[1024];
  __shared__ float q_part[8][4];
  __shared__ float s_part[8][3];

  const long long rowBase = (long long)p * N;

  // ---- stage this thread's 4 steps of each array: async global -> LDS -----
  {
    const float* gs = sp + rowBase + 4 * t;
    const float* gp = pa + rowBase + 4 * t;
    const float* gn = na + rowBase + 4 * t;
    float* ls = s_sp + 4 * t;
    float* lp = s_pa + 4 * t;
    float* ln = s_na + 4 * t;
#if HAVE_ASYNC
    __builtin_amdgcn_global_load_async_to_lds_b128(
        (AS1 v4i*)(uintptr_t)gs, (AS3 v4i*)(AS3 void*)ls, 0, 0);
    __builtin_amdgcn_global_load_async_to_lds_b128(
        (AS1 v4i*)(uintptr_t)gp, (AS3 v4i*)(AS3 void*)lp, 0, 0);
    __builtin_amdgcn_global_load_async_to_lds_b128(
        (AS1 v4i*)(uintptr_t)gn, (AS3 v4i*)(AS3 void*)ln, 0, 0);
    WAIT_ASYNC0();                 // each thread reads back only its own tile
#else
    *(v4f*)ls = *(const v4f*)gs;
    *(v4f*)lp = *(const v4f*)gp;
    *(v4f*)ln = *(const v4f*)gn;
#endif
  }
  const v4f vs = *(const v4f*)(s_sp + 4 * t);
  const v4f va = *(const v4f*)(s_pa + 4 * t);
  const v4f vn = *(const v4f*)(s_na + 4 * t);

  // ---- local quaternions and thread-local inclusive prefixes --------------
  Quat L[4];
  {
    const float pa_[4] = {va.x, va.y, va.z, va.w};
    const float na_[4] = {vn.x, vn.y, vn.z, vn.w};
    Quat run = {1.0f, 0.0f, 0.0f, 0.0f};
#pragma unroll
    for (int j = 0; j < 4; ++j) {
      float cz, sz, cy, sy;
      sincosf(0.5f * pa_[j], &sz, &cz);     // Rz(pa)
      sincosf(0.5f * na_[j], &sy, &cy);     // Ry(na)
      const Quat q = { cz * cy, -sz * sy, cz * sy, cy * sz };  // qz (x) qy
      run  = qmul(run, q);
      L[j] = run;
    }
  }

  // ---- wave-level inclusive scan of quaternion products (ordered) ---------
  Quat inc = L[3];
#pragma unroll
  for (int d = 1; d < 32; d <<= 1) {
    Quat o = qshfl_up(inc, d);
    if (lane >= d) inc = qmul(o, inc);      // earlier steps on the left
  }
  Quat excl = qshfl_up(inc, 1);
  if (lane == 0) excl = {1.0f, 0.0f, 0.0f, 0.0f};
  if (lane == 31) {
    q_part[wid][0] = inc.w; q_part[wid][1] = inc.x;
    q_part[wid][2] = inc.y; q_part[wid][3] = inc.z;
  }
  __syncthreads();

  Quat Ew = {1.0f, 0.0f, 0.0f, 0.0f};
  for (int w = 0; w < wid; ++w) {           // <=7 broadcast LDS reads
    const Quat qw = { q_part[w][0], q_part[w][1], q_part[w][2], q_part[w][3] };
    Ew = qmul(Ew, qw);
  }
  const Quat Et = qmul(Ew, excl);           // product of all earlier steps

  // ---- initial frame / position (broadcast loads) -------------------------
  const F3 E0 = { e0i[3*p], e0i[3*p+1], e0i[3*p+2] };
  const F3 E1 = { e1i[3*p], e1i[3*p+1], e1i[3*p+2] };
  const F3 E2 = { e2i[3*p], e2i[3*p+1], e2i[3*p+2] };
  const F3 X0 = { x0[3*p],  x0[3*p+1],  x0[3*p+2]  };

  // ---- per-step e0 and local inclusive displacement sums ------------------
  const float sp_[4] = {vs.x, vs.y, vs.z, vs.w};
  F3 lsum[4];
  F3 runs = {0.0f, 0.0f, 0.0f};
#pragma unroll
  for (int j = 0; j < 4; ++j) {
    const Quat P = qmul(Et, L[j]);
    const float n2   = P.w*P.w + P.x*P.x + P.y*P.y + P.z*P.z;
    const float inv2 = 2.0f / n2;           // folds |q| normalisation in
    const float r00 = 1.0f - inv2 * (P.y*P.y + P.z*P.z);
    const float r10 = inv2 * (P.x*P.y + P.w*P.z);
    const float r20 = inv2 * (P.x*P.z - P.w*P.y);
    const F3 e0 = { r00*E0.x + r10*E1.x + r20*E2.x,
                    r00*E0.y + r10*E1.y + r20*E2.y,
                    r00*E0.z + r10*E1.z + r20*E2.z };
    runs.x += sp_[j] * e0.x;
    runs.y += sp_[j] * e0.y;
    runs.z += sp_[j] * e0.z;
    lsum[j] = runs;
  }

  // ---- wave-level inclusive scan of displacement sums ---------------------
  F3 incs = runs;
#pragma unroll
  for (int d = 1; d < 32; d <<= 1) {
    F3 o = fshfl_up(incs, d);
    if (lane >= d) { incs.x += o.x; incs.y += o.y; incs.z += o.z; }
  }
  F3 excls = fshfl_up(incs, 1);
  if (lane == 0) excls = {0.0f, 0.0f, 0.0f};
  if (lane == 31) {
    s_part[wid][0] = incs.x; s_part[wid][1] = incs.y; s_part[wid][2] = incs.z;
  }
  __syncthreads();

  F3 Sw = {0.0f, 0.0f, 0.0f};
  for (int w = 0; w < wid; ++w) {
    Sw.x += s_part[w][0]; Sw.y += s_part[w][1]; Sw.z += s_part[w][2];
  }
  const F3 base = { X0.x + Sw.x + excls.x,
                    X0.y + Sw.y + excls.y,
                    X0.z + Sw.z + excls.z };

  // ---- emit 4 positions (48B contiguous, 16B aligned) as NT b128 stores ---
  float xb[12];
#pragma unroll
  for (int j = 0; j < 4; ++j) {
    xb[3*j+0] = base.x + lsum[j].x;
    xb[3*j+1] = base.y + lsum[j].y;
    xb[3*j+2] = base.z + lsum[j].z;
  }
  float* dst = X + (rowBase + 4 * t) * 3;
  const v4f a0 = {xb[0], xb[1], xb[2],  xb[3]};
  const v4f a1 = {xb[4], xb[5], xb[6],  xb[7]};
  const v4f a2 = {xb[8], xb[9], xb[10], xb[11]};
  __builtin_nontemporal_store(a0, (v4f*)dst);
  __builtin_nontemporal_store(a1, (v4f*)dst + 1);
  __builtin_nontemporal_store(a2, (v4f*)dst + 2);
}

// ---------------------------------------------------------------------------
extern "C" void kernel_launch(void* const* d_in, const int* in_sizes, int n_in,
                              void* d_out, int out_size, void* d_ws, size_t ws_size,
                              hipStream_t stream)
{
  const float* x0  = (const float*)d_in[0];
  const float* e0i = (const float*)d_in[1];
  const float* e1i = (const float*)d_in[2];
  const float* e2i = (const float*)d_in[3];
  const float* sp  = (const float*)d_in[4];
  const float* pa  = (const float*)d_in[5];
  const float* na  = (const float*)d_in[6];

  const long long B  = in_sizes[0] / 3;     // 8192
  const long long BN = in_sizes[4];         // B * N
  const int       N  = (int)(BN / B);       // 1024 (dt == 1 -> n_steps == N)

  // Output layout: ts | X | states | speeds | planar | nonplanar
  float* out      = (float*)d_out;
  float* o_ts     = out;
  float* o_X      = o_ts + N;
  float* o_states = o_X + 3 * BN;
  float* o_sp     = o_states + BN;
  float* o_pa     = o_sp + BN;
  float* o_na     = o_pa + BN;

  ParticleExplorer_64561948393590_misc<<<2048, 256, 0, stream>>>(o_ts, o_states, N, BN);

  (void)hipMemcpyAsync(o_sp, sp, sizeof(float) * BN, hipMemcpyDeviceToDevice, stream);
  (void)hipMemcpyAsync(o_pa, pa, sizeof(float) * BN, hipMemcpyDeviceToDevice, stream);
  (void)hipMemcpyAsync(o_na, na, sizeof(float) * BN, hipMemcpyDeviceToDevice, stream);

  // One block per particle; blockDim = N/4 = 256 threads (4 steps/thread).
  ParticleExplorer_64561948393590_scan<<<(int)B, N / 4, 0, stream>>>(
      x0, e0i, e1i, e2i, sp, pa, na, o_X, N);
}